// MambaBlock_76390288326722
// MI455X (gfx1250) — compile-verified
//
#include <hip/hip_runtime.h>
#include <math.h>
#include <stdint.h>

// ---------------------------------------------------------------------------
// Mamba block forward, fp32 throughout (matches reference precision).
// GEMMs: V_WMMA_F32_16X16X4_F32 (CDNA5 wave32 WMMA, fp32 in / fp32 acc).
// Big GEMMs stage A/B K-blocks into LDS, double-buffered, via the CDNA5
// async-to-LDS path (GLOBAL_LOAD_ASYNC_TO_LDS_B64, ASYNCcnt) using inline asm.
// ---------------------------------------------------------------------------

typedef float v2f __attribute__((ext_vector_type(2)));
typedef float v8f __attribute__((ext_vector_type(8)));

#define D_MODEL 1024
#define D_INNER 2048
#define D_STATE 16
#define DT_RANK 64
#define SEQ_L   2048
#define XZ_LD   (2 * D_INNER)           // 4096
#define XDBL_LD (DT_RANK + 2 * D_STATE) // 96

#if defined(__gfx1250__) && __has_builtin(__builtin_amdgcn_s_wait_asynccnt)
#define USE_ASYNC_LDS 1
#else
#define USE_ASYNC_LDS 0
#endif

// 8-byte global -> LDS copy. Async path writes LDS directly from the memory
// system (no VGPR round-trip, tracked by ASYNCcnt); fallback is reg+ds_store.
// CDNA5 ISA (GV mode): the LDS address is the LDS base plus the VDST VGPR value; the LDS
// byte offset is the low 32 bits of the flat shared-memory pointer.
__device__ __forceinline__ void copy8_g2l(const float* __restrict__ g, float* l) {
#if USE_ASYNC_LDS
  const uint32_t loff  = (uint32_t)(uintptr_t)l;
  const uint64_t gaddr = (uint64_t)(uintptr_t)g;
  asm volatile("global_load_async_to_lds_b64 %0, %1, off"
               :
               : "v"(loff), "v"(gaddr)
               : "memory");
#else
  const float2 v = *(const float2*)g;
  *(float2*)l = v;
#endif
}

__device__ __forceinline__ void stage_wait() {
#if USE_ASYNC_LDS
  __builtin_amdgcn_s_wait_asynccnt(0);  // own wave's async copies complete
#endif
}

// ---- fp32 WMMA fragment layouts (ISA 7.12.2) -------------------------------
// A (16x4): lanes 0-15 hold M=0..15 {K=0,K=1}; lanes 16-31 hold {K=2,K=3}.
// B (4x16) mirrors it with N across lanes; our "B" is a [N,K] row-major
// weight, so both fragments use identical addressing.
__device__ __forceinline__ v2f frag_ld_g(const float* __restrict__ p, int ld,
                                         int base, int k, int lane) {
  const int rc   = lane & 15;
  const int koff = (lane >> 4) << 1;
  const float* q = p + (size_t)(base + rc) * (size_t)ld + (size_t)(k + koff);
  v2f r; r.x = q[0]; r.y = q[1];
  return r;
}

#define LDS_PAD 36  // floats per staged row: 8B-aligned b64 rows, 36r%64 hits
                    // distinct multiples of 4 -> conflict-free fragment reads

__device__ __forceinline__ v2f frag_ld_l(const float* p, int row0, int kk, int lane) {
  const int rc   = lane & 15;
  const int koff = (lane >> 4) << 1;
  const float* q = p + (row0 + rc) * LDS_PAD + kk + koff;
  v2f r; r.x = q[0]; r.y = q[1];
  return r;
}

__device__ __forceinline__ float softplusf(float v) {
  return (v > 20.0f) ? v : log1pf(__expf(v));
}

// ---------------------------------------------------------------------------
// LDS-staged, double-buffered WMMA GEMM: C = A[M,K] * W[N,K]^T.
// Block: 256 threads = 8 waves; block tile 64(M) x 128(N); K-block 32.
// Each wave owns a 32x32 tile (2x2 accumulators). Requires M%64==0, N%128==0,
// K%32==0. EPILOGUE 0: plain store. EPILOGUE 1: +bias[n], softplus.
// ---------------------------------------------------------------------------
#define BM 64
#define BN 128
#define BK 32

template <int EPILOGUE>
__global__ void __launch_bounds__(256)
wmma_gemm_lds(const float* __restrict__ A, const float* __restrict__ W,
              float* __restrict__ C, const float* __restrict__ bias,
              int K, int N, int lda, int ldb, int ldc) {
  __shared__ float sA[2][BM * LDS_PAD];
  __shared__ float sB[2][BN * LDS_PAD];

  const int tid  = threadIdx.x;
  const int lane = tid & 31;
  const int wave = tid >> 5;
  const int tiles_n = N / BN;
  const int m0 = (blockIdx.x / tiles_n) * BM;
  const int n0 = (blockIdx.x % tiles_n) * BN;
  const int wm = (wave & 1) * 32;   // wave's M offset inside block tile
  const int wn = (wave >> 1) * 32;  // wave's N offset inside block tile

  v8f acc00 = {}; v8f acc01 = {}; v8f acc10 = {}; v8f acc11 = {};

  // Stage one K-block (BM x BK of A, BN x BK of W) into LDS buffer `buf`,
  // float2 granularity, fully coalesced along K.
  auto stage = [&](int buf, int k0) {
    #pragma unroll
    for (int it = 0; it < (BM * BK / 2) / 256; ++it) {   // 4 items/thread
      const int i = tid + it * 256;
      const int row = i >> 4, q = i & 15;
      copy8_g2l(A + (size_t)(m0 + row) * lda + k0 + q * 2,
                &sA[buf][row * LDS_PAD + q * 2]);
    }
    #pragma unroll
    for (int it = 0; it < (BN * BK / 2) / 256; ++it) {   // 8 items/thread
      const int i = tid + it * 256;
      const int row = i >> 4, q = i & 15;
      copy8_g2l(W + (size_t)(n0 + row) * ldb + k0 + q * 2,
                &sB[buf][row * LDS_PAD + q * 2]);
    }
  };

  stage(0, 0);
  const int nkb = K / BK;
  for (int kb = 0; kb < nkb; ++kb) {
    stage_wait();        // own wave's async copies into `cur` complete
    __syncthreads();     // every wave's copies visible; prev compute done
    if (kb + 1 < nkb) stage((kb + 1) & 1, (kb + 1) * BK);

    const int buf = kb & 1;
    const float* pa = &sA[buf][0];
    const float* pb = &sB[buf][0];
    #pragma unroll
    for (int kk = 0; kk < BK; kk += 4) {
      v2f a0 = frag_ld_l(pa, wm,      kk, lane);
      v2f a1 = frag_ld_l(pa, wm + 16, kk, lane);
      v2f b0 = frag_ld_l(pb, wn,      kk, lane);
      v2f b1 = frag_ld_l(pb, wn + 16, kk, lane);
      acc00 = __builtin_amdgcn_wmma_f32_16x16x4_f32(false, a0, false, b0, (short)0, acc00, false, false);
      acc01 = __builtin_amdgcn_wmma_f32_16x16x4_f32(false, a0, false, b1, (short)0, acc01, false, false);
      acc10 = __builtin_amdgcn_wmma_f32_16x16x4_f32(false, a1, false, b0, (short)0, acc10, false, false);
      acc11 = __builtin_amdgcn_wmma_f32_16x16x4_f32(false, a1, false, b1, (short)0, acc11, false, false);
    }
  }

  // C/D layout: VGPR r, lanes 0-15 -> M=r, lanes 16-31 -> M=8+r; N = lane%16.
  const int col = lane & 15;
  const int mo  = (lane >> 4) << 3;
  auto store_tile = [&](const v8f& acc, int mb, int nb) {
    #pragma unroll
    for (int r = 0; r < 8; ++r) {
      float v = acc[r];
      const int cidx = n0 + nb + col;
      if (EPILOGUE == 1) { v += bias[cidx]; v = softplusf(v); }
      C[(size_t)(m0 + mb + mo + r) * (size_t)ldc + cidx] = v;
    }
  };
  store_tile(acc00, wm,      wn);
  store_tile(acc01, wm,      wn + 16);
  store_tile(acc10, wm + 16, wn);
  store_tile(acc11, wm + 16, wn + 16);
}

// ---------------------------------------------------------------------------
// Direct-load WMMA GEMM for the small x_proj GEMM (N=96, not a 128 multiple).
// ---------------------------------------------------------------------------
__global__ void __launch_bounds__(128)
wmma_gemm_direct(const float* __restrict__ A, const float* __restrict__ W,
                 float* __restrict__ C, int K, int N,
                 int lda, int ldb, int ldc, int total_tiles) {
  const int lane = threadIdx.x & 31;
  const int wave = threadIdx.x >> 5;
  const int tiles_n = N >> 5;
  const int tile = blockIdx.x * 4 + wave;
  if (tile >= total_tiles) return;
  const int m0 = (tile / tiles_n) << 5;
  const int n0 = (tile % tiles_n) << 5;

  v8f acc00 = {}; v8f acc01 = {}; v8f acc10 = {}; v8f acc11 = {};
  #pragma unroll 4
  for (int k = 0; k < K; k += 4) {
    v2f a0 = frag_ld_g(A, lda, m0,      k, lane);
    v2f a1 = frag_ld_g(A, lda, m0 + 16, k, lane);
    v2f b0 = frag_ld_g(W, ldb, n0,      k, lane);
    v2f b1 = frag_ld_g(W, ldb, n0 + 16, k, lane);
    acc00 = __builtin_amdgcn_wmma_f32_16x16x4_f32(false, a0, false, b0, (short)0, acc00, false, false);
    acc01 = __builtin_amdgcn_wmma_f32_16x16x4_f32(false, a0, false, b1, (short)0, acc01, false, false);
    acc10 = __builtin_amdgcn_wmma_f32_16x16x4_f32(false, a1, false, b0, (short)0, acc10, false, false);
    acc11 = __builtin_amdgcn_wmma_f32_16x16x4_f32(false, a1, false, b1, (short)0, acc11, false, false);
  }

  const int col = lane & 15;
  const int mo  = (lane >> 4) << 3;
  auto store_tile = [&](const v8f& acc, int mb, int nb) {
    #pragma unroll
    for (int r = 0; r < 8; ++r)
      C[(size_t)(mb + mo + r) * (size_t)ldc + nb + col] = acc[r];
  };
  store_tile(acc00, m0,      n0);
  store_tile(acc01, m0,      n0 + 16);
  store_tile(acc10, m0 + 16, n0);
  store_tile(acc11, m0 + 16, n0 + 16);
}

// ---- causal depthwise conv (width 4) + bias + SiLU -------------------------
__global__ void __launch_bounds__(256)
conv_silu_kernel(const float* __restrict__ xz, const float* __restrict__ w,
                 const float* __restrict__ b, float* __restrict__ xconv) {
  const int idx = blockIdx.x * blockDim.x + threadIdx.x;   // over L*D_INNER
  const int d = idx & (D_INNER - 1);
  const int l = idx >> 11;                                 // / D_INNER
  float acc = b[d];
  #pragma unroll
  for (int t = 0; t < 4; ++t) {
    const int ll = l - 3 + t;
    if (ll >= 0) acc += xz[(size_t)ll * XZ_LD + d] * w[d * 4 + t];
  }
  xconv[idx] = acc / (1.0f + __expf(-acc));                // silu
}

// ---- selective scan: one thread per channel, 16 states in registers --------
#define SCAN_CHUNK 128
__global__ void __launch_bounds__(256)
scan_kernel(const float* __restrict__ dt, const float* __restrict__ xconv,
            const float* __restrict__ xdbl, const float* __restrict__ xz,
            const float* __restrict__ A_log, const float* __restrict__ Dp,
            float* __restrict__ yz) {
  __shared__ float sB[SCAN_CHUNK][D_STATE];
  __shared__ float sC[SCAN_CHUNK][D_STATE];
  const int d = blockIdx.x * blockDim.x + threadIdx.x;     // channel

  float a[D_STATE], h[D_STATE];
  #pragma unroll
  for (int n = 0; n < D_STATE; ++n) {
    a[n] = -__expf(A_log[d * D_STATE + n]);
    h[n] = 0.0f;
  }
  const float Dd = Dp[d];

  for (int l0 = 0; l0 < SEQ_L; l0 += SCAN_CHUNK) {
    // stage broadcast B/C vectors (shared by all channels) into LDS
    for (int i = threadIdx.x; i < SCAN_CHUNK * 2 * D_STATE; i += blockDim.x) {
      const int ll = i >> 5;
      const int j  = i & 31;
      const float v = xdbl[(size_t)(l0 + ll) * XDBL_LD + DT_RANK + j];
      if (j < D_STATE) sB[ll][j] = v; else sC[ll][j - D_STATE] = v;
    }
    __syncthreads();

    for (int li = 0; li < SCAN_CHUNK; ++li) {
      const int l = l0 + li;
      const float dtv = dt[(size_t)l * D_INNER + d];
      const float xc  = xconv[(size_t)l * D_INNER + d];
      const float dbx = dtv * xc;
      float y = 0.0f;
      #pragma unroll
      for (int n = 0; n < D_STATE; ++n) {
        const float dA = __expf(dtv * a[n]);
        h[n] = dA * h[n] + dbx * sB[li][n];
        y += h[n] * sC[li][n];
      }
      y += xc * Dd;
      const float zv = xz[(size_t)l * XZ_LD + D_INNER + d];
      const float sz = zv / (1.0f + __expf(-zv));          // silu(z)
      yz[(size_t)l * D_INNER + d] = y * sz;
    }
    __syncthreads();
  }
}

// ---------------------------------------------------------------------------
extern "C" void kernel_launch(void* const* d_in, const int* in_sizes, int n_in,
                              void* d_out, int out_size, void* d_ws, size_t ws_size,
                              hipStream_t stream) {
  const float* x          = (const float*)d_in[0];
  const float* in_proj_w  = (const float*)d_in[1];
  const float* conv_w     = (const float*)d_in[2];
  const float* conv_b     = (const float*)d_in[3];
  const float* x_proj_w   = (const float*)d_in[4];
  const float* dt_proj_w  = (const float*)d_in[5];
  const float* dt_proj_b  = (const float*)d_in[6];
  const float* A_log      = (const float*)d_in[7];
  const float* D_param    = (const float*)d_in[8];
  const float* out_proj_w = (const float*)d_in[9];
  float* out = (float*)d_out;

  float* ws    = (float*)d_ws;
  float* xz    = ws;                                  // [2048, 4096]
  float* xconv = xz    + (size_t)SEQ_L * XZ_LD;       // [2048, 2048]
  float* xdbl  = xconv + (size_t)SEQ_L * D_INNER;     // [2048, 96]
  float* dt    = xdbl  + (size_t)SEQ_L * XDBL_LD;     // [2048, 2048]
  float* yz    = dt    + (size_t)SEQ_L * D_INNER;     // [2048, 2048]

  // 1) xz = x @ in_proj_w^T   (2048 x 4096, K=1024)
  wmma_gemm_lds<0><<<(SEQ_L / BM) * (XZ_LD / BN), 256, 0, stream>>>(
      x, in_proj_w, xz, nullptr, D_MODEL, XZ_LD, D_MODEL, D_MODEL, XZ_LD);

  // 2) x_conv = silu(depthwise_conv(x_inner) + b)
  conv_silu_kernel<<<(SEQ_L * D_INNER) / 256, 256, 0, stream>>>(xz, conv_w, conv_b, xconv);

  // 3) x_dbl = x_conv @ x_proj_w^T   (2048 x 96, K=2048) -- N=96, direct path
  {
    const int tiles = (SEQ_L / 32) * (XDBL_LD / 32);  // 192
    wmma_gemm_direct<<<(tiles + 3) / 4, 128, 0, stream>>>(
        xconv, x_proj_w, xdbl, D_INNER, XDBL_LD, D_INNER, D_INNER, XDBL_LD, tiles);
  }

  // 4) dt = softplus(x_dbl[:, :64] @ dt_proj_w^T + dt_proj_b)  (2048 x 2048, K=64)
  wmma_gemm_lds<1><<<(SEQ_L / BM) * (D_INNER / BN), 256, 0, stream>>>(
      xdbl, dt_proj_w, dt, dt_proj_b, DT_RANK, D_INNER, XDBL_LD, DT_RANK, D_INNER);

  // 5) selective scan + D skip + silu(z) gating -> yz
  scan_kernel<<<D_INNER / 256, 256, 0, stream>>>(dt, xconv, xdbl, xz, A_log, D_param, yz);

  // 6) out = yz @ out_proj_w^T   (2048 x 1024, K=2048)
  wmma_gemm_lds<0><<<(SEQ_L / BM) * (D_MODEL / BN), 256, 0, stream>>>(
      yz, out_proj_w, out, nullptr, D_INNER, D_MODEL, D_INNER, D_INNER, D_MODEL);
}